// GIN_R_34376918237439
// MI455X (gfx1250) — compile-verified
//
#include <hip/hip_runtime.h>
#include <hip/hip_bf16.h>

#define N_NODES   100000
#define N_EDGES   1000000
#define DIM       64
#define N_GRAPHS  512
#define N_CLS     10
#define BN_EPS    1e-5f
#define LDS_STRIDE 68   // 64 + 4 pad: bank = (4*row + col) % 64 -> conflict-free columns

typedef float v2f __attribute__((ext_vector_type(2)));
typedef float v8f __attribute__((ext_vector_type(8)));

// ---------------------------------------------------------------------------
// bufA = src  (vectorized copy, N_NODES*64 floats = 1.6M float4)
// ---------------------------------------------------------------------------
__global__ __launch_bounds__(256)
void gin_copy_kernel(float4* __restrict__ dst, const float4* __restrict__ src, int n4) {
    int t = blockIdx.x * blockDim.x + threadIdx.x;
    if (t < n4) dst[t] = src[t];
}

// ---------------------------------------------------------------------------
// acc[dst[e]] += h[src[e]]   (16 threads per edge, float4 gather, fp32 atomics)
// ---------------------------------------------------------------------------
__global__ __launch_bounds__(256)
void gin_scatter_kernel(float* __restrict__ acc, const float* __restrict__ h,
                        const int* __restrict__ src, const int* __restrict__ dst) {
    int t = blockIdx.x * blockDim.x + threadIdx.x;
    int e = t >> 4;
    if (e >= N_EDGES) return;
    int q = t & 15;
    int s = src[e];
    int d = dst[e];
    float4 v = reinterpret_cast<const float4*>(h + (size_t)s * DIM)[q];
    float* dp = acc + (size_t)d * DIM + q * 4;
    atomicAdd(dp + 0, v.x);
    atomicAdd(dp + 1, v.y);
    atomicAdd(dp + 2, v.z);
    atomicAdd(dp + 3, v.w);
}

// ---------------------------------------------------------------------------
// Fused GIN MLP: out = [ReLU]( Linear_b( ReLU( BN( Linear_a(in) ) ) ) )
// 16 rows per block, 4 waves, each wave owns one 16x16 output column tile.
// Both GEMMs run on V_WMMA_F32_16X16X4_F32 (full fp32, K stepped by 4).
// ---------------------------------------------------------------------------
template<bool FINAL_RELU>
__global__ __launch_bounds__(128)
void gin_mlp_kernel(const float* __restrict__ in, float* __restrict__ out,
                    const float* __restrict__ Wa, const float* __restrict__ ba,
                    const float* __restrict__ gma, const float* __restrict__ bta,
                    const float* __restrict__ mea, const float* __restrict__ vra,
                    const float* __restrict__ Wb, const float* __restrict__ bb) {
    __shared__ float inT [16 * LDS_STRIDE];
    __shared__ float midT[16 * LDS_STRIDE];

    const int t = threadIdx.x;
    const int rowBase = blockIdx.x * 16;

    // Stage the 16x64 input tile into LDS (each thread: 8 floats).
    {
        int r = t >> 3;           // 0..15
        int c = (t & 7) * 8;      // 0,8,...,56
        const float4* g = reinterpret_cast<const float4*>(in + (size_t)(rowBase + r) * DIM + c);
        float4 v0 = g[0];
        float4 v1 = g[1];
        *reinterpret_cast<float4*>(&inT[r * LDS_STRIDE + c])     = v0;
        *reinterpret_cast<float4*>(&inT[r * LDS_STRIDE + c + 4]) = v1;
    }
    __syncthreads();

    const int lane = t & 31;
    const int wv   = t >> 5;        // wave id 0..3 -> column tile
    const int half = lane >> 4;     // 0: K=k0..k0+1, 1: K=k0+2..k0+3
    const int lm   = lane & 15;     // row (A) / column-in-tile (B,C,D)
    const int n    = wv * 16 + lm;  // global output column 0..63

    // ---- GEMM1: inT(16x64) @ Wa(64x64), bias folded into accumulator ----
    v8f acc;
    {
        float bias = ba[n];
        #pragma unroll
        for (int i = 0; i < 8; ++i) acc[i] = bias;
    }
    #pragma unroll
    for (int k0 = 0; k0 < DIM; k0 += 4) {
        const int kk = k0 + 2 * half;
        v2f a, b;
        a.x = inT[lm * LDS_STRIDE + kk];
        a.y = inT[lm * LDS_STRIDE + kk + 1];
        b.x = Wa[(size_t)kk * DIM + n];
        b.y = Wa[(size_t)(kk + 1) * DIM + n];
        acc = __builtin_amdgcn_wmma_f32_16x16x4_f32(false, a, false, b, (short)0, acc,
                                                    false, false);
    }

    // ---- BN (eval) + ReLU, stash 16x64 intermediate in LDS ----
    {
        float sc = gma[n] * rsqrtf(vra[n] + BN_EPS);
        float sh = bta[n] - mea[n] * sc;
        #pragma unroll
        for (int i = 0; i < 8; ++i) {
            int r = i + 8 * half;   // C/D layout: VGPR i holds rows i (lanes 0-15), i+8 (16-31)
            midT[r * LDS_STRIDE + n] = fmaxf(acc[i] * sc + sh, 0.0f);
        }
    }
    __syncthreads();

    // ---- GEMM2: midT(16x64) @ Wb(64x64) ----
    v8f acc2;
    {
        float bias = bb[n];
        #pragma unroll
        for (int i = 0; i < 8; ++i) acc2[i] = bias;
    }
    #pragma unroll
    for (int k0 = 0; k0 < DIM; k0 += 4) {
        const int kk = k0 + 2 * half;
        v2f a, b;
        a.x = midT[lm * LDS_STRIDE + kk];
        a.y = midT[lm * LDS_STRIDE + kk + 1];
        b.x = Wb[(size_t)kk * DIM + n];
        b.y = Wb[(size_t)(kk + 1) * DIM + n];
        acc2 = __builtin_amdgcn_wmma_f32_16x16x4_f32(false, a, false, b, (short)0, acc2,
                                                     false, false);
    }

    #pragma unroll
    for (int i = 0; i < 8; ++i) {
        int r = i + 8 * half;
        float v = acc2[i];
        if (FINAL_RELU) v = fmaxf(v, 0.0f);
        out[(size_t)(rowBase + r) * DIM + n] = v;
    }
}

// ---------------------------------------------------------------------------
// Zero pooled sums + counts (scratch is poisoned, must re-zero every call)
// ---------------------------------------------------------------------------
__global__ __launch_bounds__(256)
void gin_zero_kernel(float* __restrict__ p, int n) {
    int t = blockIdx.x * blockDim.x + threadIdx.x;
    if (t < n) p[t] = 0.0f;
}

// ---------------------------------------------------------------------------
// Global mean-pool (accumulate): sums[batch[i]] += h[i]; cnts[batch[i]] += 1
// ---------------------------------------------------------------------------
__global__ __launch_bounds__(256)
void gin_pool_kernel(const float* __restrict__ h, const int* __restrict__ batch,
                     float* __restrict__ sums, float* __restrict__ cnts) {
    int t = blockIdx.x * blockDim.x + threadIdx.x;
    int node = t >> 4;
    if (node >= N_NODES) return;
    int q = t & 15;
    int g = batch[node];
    float4 v = reinterpret_cast<const float4*>(h + (size_t)node * DIM)[q];
    float* dp = sums + (size_t)g * DIM + q * 4;
    atomicAdd(dp + 0, v.x);
    atomicAdd(dp + 1, v.y);
    atomicAdd(dp + 2, v.z);
    atomicAdd(dp + 3, v.w);
    if (q == 0) atomicAdd(cnts + g, 1.0f);
}

// ---------------------------------------------------------------------------
// out[g] = (sums[g]/max(cnt,1)) @ wc + bc      (512 graphs x 10 classes)
// ---------------------------------------------------------------------------
__global__ __launch_bounds__(128)
void gin_classify_kernel(const float* __restrict__ sums, const float* __restrict__ cnts,
                         const float* __restrict__ wc, const float* __restrict__ bc,
                         float* __restrict__ out) {
    int g = blockIdx.x * blockDim.x + threadIdx.x;
    if (g >= N_GRAPHS) return;
    float inv = 1.0f / fmaxf(cnts[g], 1.0f);
    float p[DIM];
    #pragma unroll
    for (int k = 0; k < DIM; ++k) p[k] = sums[(size_t)g * DIM + k] * inv;
    #pragma unroll
    for (int c = 0; c < N_CLS; ++c) {
        float a = bc[c];
        #pragma unroll
        for (int k = 0; k < DIM; ++k) a = fmaf(p[k], wc[(size_t)k * N_CLS + c], a);
        out[(size_t)g * N_CLS + c] = a;
    }
}

// ---------------------------------------------------------------------------
extern "C" void kernel_launch(void* const* d_in, const int* in_sizes, int n_in,
                              void* d_out, int out_size, void* d_ws, size_t ws_size,
                              hipStream_t stream) {
    const float* x     = (const float*)d_in[0];
    const int*   src   = (const int*)d_in[1];              // edge_index[0]
    const int*   dst   = src + N_EDGES;                    // edge_index[1]
    const int*   batch = (const int*)d_in[2];
    const float* w0a = (const float*)d_in[3],  *b0a = (const float*)d_in[4];
    const float* g0  = (const float*)d_in[5],  *be0 = (const float*)d_in[6];
    const float* m0  = (const float*)d_in[7],  *v0  = (const float*)d_in[8];
    const float* w0b = (const float*)d_in[9],  *b0b = (const float*)d_in[10];
    const float* w1a = (const float*)d_in[11], *b1a = (const float*)d_in[12];
    const float* g1  = (const float*)d_in[13], *be1 = (const float*)d_in[14];
    const float* m1  = (const float*)d_in[15], *v1  = (const float*)d_in[16];
    const float* w1b = (const float*)d_in[17], *b1b = (const float*)d_in[18];
    const float* wc  = (const float*)d_in[19], *bc  = (const float*)d_in[20];
    float* out = (float*)d_out;

    // Workspace layout
    float* bufA = (float*)d_ws;                  // N_NODES*64: h + agg
    float* bufB = bufA + (size_t)N_NODES * DIM;  // N_NODES*64: conv output
    float* sums = bufB + (size_t)N_NODES * DIM;  // 512*64
    float* cnts = sums + (size_t)N_GRAPHS * DIM; // 512

    const int n4       = N_NODES * DIM / 4;                 // float4 copy count
    const int copyGrid = (n4 + 255) / 256;
    const int edgeGrid = (N_EDGES * 16 + 255) / 256;
    const int nodeGrid = (N_NODES * 16 + 255) / 256;
    const int mlpGrid  = N_NODES / 16;                      // 100000/16 = 6250 exact
    const int poolN    = N_GRAPHS * DIM + N_GRAPHS;

    // ---- conv0: bufA = x + scatter_add(x); bufB = relu(mlp0(bufA)) ----
    gin_copy_kernel<<<copyGrid, 256, 0, stream>>>((float4*)bufA, (const float4*)x, n4);
    gin_scatter_kernel<<<edgeGrid, 256, 0, stream>>>(bufA, x, src, dst);
    gin_mlp_kernel<true><<<mlpGrid, 128, 0, stream>>>(bufA, bufB,
        w0a, b0a, g0, be0, m0, v0, w0b, b0b);

    // ---- conv1: bufA = bufB + scatter_add(bufB); bufB = mlp1(bufA) ----
    gin_copy_kernel<<<copyGrid, 256, 0, stream>>>((float4*)bufA, (const float4*)bufB, n4);
    gin_scatter_kernel<<<edgeGrid, 256, 0, stream>>>(bufA, bufB, src, dst);
    gin_mlp_kernel<false><<<mlpGrid, 128, 0, stream>>>(bufA, bufB,
        w1a, b1a, g1, be1, m1, v1, w1b, b1b);

    // ---- global mean pool + classifier ----
    gin_zero_kernel<<<(poolN + 255) / 256, 256, 0, stream>>>(sums, poolN);
    gin_pool_kernel<<<nodeGrid, 256, 0, stream>>>(bufB, batch, sums, cnts);
    gin_classify_kernel<<<(N_GRAPHS + 127) / 128, 128, 0, stream>>>(sums, cnts, wc, bc, out);
}